// ChainLoss_67319317397938
// MI455X (gfx1250) — compile-verified
//
#include <hip/hip_runtime.h>
#include <stdint.h>

#define TT 400
#define DD 2048
#define BB 32

// ---------------- den graph config (shared edges) ----------------
#define S_DEN 2000
#define E_DEN 60000
#define DEN_BSUB 4            // batch columns per workgroup (lanes 0..3 within 4-lane group)
#define DEN_BLOCKS 8          // 8 * 4 = 32 batch
#define DEN_CH 512            // edges per staged chunk
#define DEN_NCH ((E_DEN + DEN_CH - 1) / DEN_CH)   // 118

// den LDS layout (float indices into dynamic smem)
#define DEN_OFF_A     0                      // S_DEN*4 = 8000
#define DEN_OFF_B     (S_DEN * DEN_BSUB)     // 8000
#define DEN_OFF_LK    (2 * S_DEN * DEN_BSUB) // 16000
#define DEN_OFF_RED   (DEN_OFF_LK + S_DEN)   // 18000 (128 floats)
#define DEN_OFF_ASUM  (DEN_OFF_RED + 128)    // 18128 (4)
#define DEN_OFF_LOGZ  (DEN_OFF_ASUM + 4)     // 18132 (4)
#define DEN_OFF_LEN   (DEN_OFF_LOGZ + 4)     // 18136 (4 ints)
#define DEN_OFF_STAGE 18144                  // 16B aligned; 2*DEN_CH*4 floats
#define DEN_LDS_FLOATS (DEN_OFF_STAGE + 2 * DEN_CH * 4)   // 22240 -> 88,960 B
static_assert((DEN_OFF_STAGE * 4) % 16 == 0, "stage must be 16B aligned");

// ---------------- num graph config (per-utterance edges) ----------------
#define S_NUM 512
#define E_NUM 2048
#define NUM_BSUB 8
#define NUM_BLOCKS 4

#define NUM_OFF_A     0
#define NUM_OFF_B     (S_NUM * NUM_BSUB)       // 4096
#define NUM_OFF_LK    (2 * S_NUM * NUM_BSUB)   // 8192
#define NUM_OFF_RED   (NUM_OFF_LK + S_NUM * NUM_BSUB)  // 12288 (256 floats)
#define NUM_OFF_ASUM  (NUM_OFF_RED + 256)      // 12544 (8)
#define NUM_OFF_LOGZ  (NUM_OFF_ASUM + 8)       // 12552 (8)
#define NUM_OFF_LEN   (NUM_OFF_LOGZ + 8)       // 12560 (8 ints)
#define NUM_LDS_FLOATS (NUM_OFF_LEN + 8)       // 12568 -> 50,272 B

// ---------------- CDNA5 async-to-LDS helpers ----------------
__device__ __forceinline__ uint32_t lds_off_of(const void* p, const void* base) {
  // dynamic LDS starts right after static LDS
  return __builtin_amdgcn_groupstaticsize() +
         (uint32_t)((const char*)p - (const char*)base);
}
__device__ __forceinline__ void async_ld_b32(uint32_t lds_byte, const void* g) {
  // GV mode: 64-bit vaddr, saddr = off. Tracked by ASYNCcnt.
  asm volatile("global_load_async_to_lds_b32 %0, %1, off"
               :: "v"(lds_byte), "v"((unsigned long long)(uintptr_t)g)
               : "memory");
}
__device__ __forceinline__ void wait_async0() {
  asm volatile("s_wait_asynccnt 0" ::: "memory");
}
__device__ __forceinline__ void wait_ds0() {
  asm volatile("s_wait_dscnt 0" ::: "memory");
}

// ---------------- kernel 1: xw[t][d][b] = exp(clip(x[b][t][d])) ----------------
__global__ __launch_bounds__(1024) void k_exp_tr(const float* __restrict__ x,
                                                 float* __restrict__ xw) {
  __shared__ float tile[32][33];
  const int tx = threadIdx.x, ty = threadIdx.y;
  const int blk = blockIdx.x;
  const int t  = blk >> 6;          // DD/32 = 64 tiles per frame
  const int d0 = (blk & 63) << 5;
  float v = x[((size_t)ty * TT + t) * DD + d0 + tx];   // b=ty, coalesced in d
  v = fminf(fmaxf(v, -30.0f), 30.0f);
  tile[ty][tx] = expf(v);
  __syncthreads();
  // write [t][d0+ty][b=tx], coalesced in b
  xw[((size_t)t * DD + d0 + ty) * BB + tx] = tile[tx][ty];
}

// ---------------- kernel 2: transpose num edge lists [B][E] -> [E][B] ----------------
__global__ void k_tr_edges(const int* __restrict__ ni, const int* __restrict__ no,
                           const int* __restrict__ np, const float* __restrict__ nq,
                           int* __restrict__ oi, int* __restrict__ oo,
                           int* __restrict__ op, float* __restrict__ oq) {
  int i = blockIdx.x * blockDim.x + threadIdx.x;
  if (i >= E_NUM * BB) return;
  int b = i & 31, e = i >> 5;
  int src = b * E_NUM + e;
  oi[i] = ni[src];
  oo[i] = no[src];
  op[i] = np[src];
  oq[i] = nq[src];
}

// ---------------- kernel 3: den forward (shared graph, async-staged edges) ----------------
__global__ __launch_bounds__(1024, 1) void k_den(
    const float* __restrict__ xw,
    const int* __restrict__ ein, const int* __restrict__ eout,
    const int* __restrict__ epdf, const float* __restrict__ eprob,
    const float* __restrict__ leaky, const float* __restrict__ finalv,
    const int* __restrict__ lengths, float* __restrict__ objf) {
  extern __shared__ float smem[];
  const int tid  = threadIdx.x;
  const int g    = blockIdx.x;          // 0..7 -> batch columns [4g, 4g+4)
  const int b4   = tid & 3;
  const int slot = tid >> 2;            // 0..255
  const int lane = tid & 31, wv = tid >> 5;
  const float coeff = 1e-5f;

  float* A     = smem + DEN_OFF_A;
  float* Bf    = smem + DEN_OFF_B;
  float* lk    = smem + DEN_OFF_LK;
  float* red   = smem + DEN_OFF_RED;
  float* asumv = smem + DEN_OFF_ASUM;
  float* logz  = smem + DEN_OFF_LOGZ;
  int*   len   = (int*)(smem + DEN_OFF_LEN);
  float* stage = smem + DEN_OFF_STAGE;  // AoS: [buf][edge][4] (in,out,pdf,prob)

  for (int i = tid; i < S_DEN; i += 1024) lk[i] = leaky[i];
  if (tid < 4) { len[tid] = lengths[g * DEN_BSUB + tid]; logz[tid] = 0.0f; }
  __syncthreads();
  for (int i = tid; i < S_DEN * DEN_BSUB; i += 1024) {
    int s = i >> 2;
    A[i] = (s == 0 ? 1.0f : 0.0f) + coeff * lk[s];
  }

  // prime chunk 0 into staging buffer 0 (async)
  for (int r = 0; r < 2; r++) {
    int idx = tid + r * 1024;               // 0..2047 dwords of the chunk
    int arr = idx >> 9, off = idx & 511;
    int e = off;
    if (e < E_DEN) {
      const void* src = (arr == 0) ? (const void*)(ein + e)
                      : (arr == 1) ? (const void*)(eout + e)
                      : (arr == 2) ? (const void*)(epdf + e)
                                   : (const void*)(eprob + e);
      async_ld_b32(lds_off_of(stage + off * 4 + arr, smem), src);
    }
  }

  const float* xwg = xw + g * DEN_BSUB;     // column base for this block
  int cur = 0;
  for (int t = 0; t < TT; t++) {
    for (int i = tid; i < S_DEN * DEN_BSUB; i += 1024) Bf[i] = 0.0f;
    const float* xrow = xwg + (size_t)t * DD * BB;
    for (int c = 0; c < DEN_NCH; c++) {
      wait_async0();           // my async copies into buffer `cur` are done
      __syncthreads();         // everyone's are done; zeroed Bf visible too
      // issue next chunk into buffer cur^1 (chunk 0 again at frame end)
      {
        int cn = (c + 1 < DEN_NCH) ? (c + 1) : 0;
        for (int r = 0; r < 2; r++) {
          int idx = tid + r * 1024;
          int arr = idx >> 9, off = idx & 511;
          int e = cn * DEN_CH + off;
          if (e < E_DEN) {
            const void* src = (arr == 0) ? (const void*)(ein + e)
                            : (arr == 1) ? (const void*)(eout + e)
                            : (arr == 2) ? (const void*)(epdf + e)
                                         : (const void*)(eprob + e);
            async_ld_b32(lds_off_of(stage + ((cur ^ 1) * DEN_CH + off) * 4 + arr, smem), src);
          }
        }
      }
      // compute chunk c: 256 edge slots x 2
      const float* sb = stage + cur * DEN_CH * 4;
      #pragma unroll
      for (int j = 0; j < 2; j++) {
        int el = slot + j * 256;
        int e  = c * DEN_CH + el;
        if (e < E_DEN) {
          uint4 pk = *(const uint4*)(sb + el * 4);   // ds_load_b128: in,out,pdf,prob
          int   in_s  = (int)pk.x;
          int   out_s = (int)pk.y;
          int   pdf   = (int)pk.z;
          float prob  = __uint_as_float(pk.w);
          float a  = A[in_s * DEN_BSUB + b4];
          float xv = xrow[pdf * BB + b4];
          float v  = a * prob * xv;
          __hip_atomic_fetch_add(&Bf[out_s * DEN_BSUB + b4], v,
                                 __ATOMIC_RELAXED, __HIP_MEMORY_SCOPE_WORKGROUP);
        }
      }
      cur ^= 1;
    }
    wait_ds0();
    __syncthreads();
    // asum[b] = sum_s Bf[s][b]
    float p = 0.0f;
    #pragma unroll
    for (int k = 0; k < 8; k++) {
      int s = slot + k * 256;
      if (s < S_DEN) p += Bf[s * DEN_BSUB + b4];
    }
    p += __shfl_xor(p, 4, 32);
    p += __shfl_xor(p, 8, 32);
    p += __shfl_xor(p, 16, 32);
    if (lane < 4) red[wv * 4 + lane] = p;
    __syncthreads();
    if (tid < 4) {
      float s = 0.0f;
      for (int w = 0; w < 32; w++) s += red[w * 4 + tid];
      asumv[tid] = s;
      if (t < len[tid]) logz[tid] += logf(s);
    }
    __syncthreads();
    // commit: adash = (anew + coeff*leaky*asum)/asum ; keep old alpha if masked
    {
      float as  = asumv[b4];
      float inv = 1.0f / as;
      bool  m   = (t < len[b4]);
      #pragma unroll
      for (int k = 0; k < 8; k++) {
        int s = slot + k * 256;
        if (s < S_DEN) {
          int i = s * DEN_BSUB + b4;
          float ad = (Bf[i] + coeff * lk[s] * as) * inv;
          Bf[i] = m ? ad : A[i];
        }
      }
    }
    __syncthreads();
    float* tmp = A; A = Bf; Bf = tmp;
  }
  // objf[b] = logz[b] + log(sum_s alpha[s][b]*final[s])
  __syncthreads();
  for (int i = tid; i < S_DEN; i += 1024) lk[i] = finalv[i];
  __syncthreads();
  float p = 0.0f;
  for (int k = 0; k < 8; k++) {
    int s = slot + k * 256;
    if (s < S_DEN) p += A[s * DEN_BSUB + b4] * lk[s];
  }
  p += __shfl_xor(p, 4, 32);
  p += __shfl_xor(p, 8, 32);
  p += __shfl_xor(p, 16, 32);
  if (lane < 4) red[wv * 4 + lane] = p;
  __syncthreads();
  if (tid < 4) {
    float s = 0.0f;
    for (int w = 0; w < 32; w++) s += red[w * 4 + tid];
    objf[g * DEN_BSUB + tid] = logz[tid] + logf(s);
  }
}

// ---------------- kernel 4: num forward (per-utterance graphs) ----------------
__global__ __launch_bounds__(1024, 1) void k_num(
    const float* __restrict__ xw,
    const int* __restrict__ ein, const int* __restrict__ eout,     // [E][32]
    const int* __restrict__ epdf, const float* __restrict__ eprob, // [E][32]
    const float* __restrict__ leaky, const float* __restrict__ finalv, // [B][S]
    const int* __restrict__ lengths, float* __restrict__ objf) {
  extern __shared__ float smem[];
  const int tid  = threadIdx.x;
  const int g    = blockIdx.x;            // 0..3 -> batch columns [8g, 8g+8)
  const int b8   = tid & 7;
  const int slot = tid >> 3;              // 0..127
  const int lane = tid & 31, wv = tid >> 5;
  const int cb   = g * NUM_BSUB + b8;     // global batch column
  const float coeff = 1e-20f;

  float* A     = smem + NUM_OFF_A;
  float* Bf    = smem + NUM_OFF_B;
  float* lk    = smem + NUM_OFF_LK;       // per-batch leaky [s][b8]
  float* red   = smem + NUM_OFF_RED;
  float* asumv = smem + NUM_OFF_ASUM;
  float* logz  = smem + NUM_OFF_LOGZ;
  int*   len   = (int*)(smem + NUM_OFF_LEN);

  for (int i = tid; i < S_NUM * NUM_BSUB; i += 1024) {
    int s = i >> 3, b = i & 7;
    lk[i] = leaky[(size_t)(g * NUM_BSUB + b) * S_NUM + s];
  }
  if (tid < 8) { len[tid] = lengths[g * NUM_BSUB + tid]; logz[tid] = 0.0f; }
  __syncthreads();
  for (int i = tid; i < S_NUM * NUM_BSUB; i += 1024) {
    int s = i >> 3;
    A[i] = (s == 0 ? 1.0f : 0.0f) + coeff * lk[i];
  }
  __syncthreads();

  for (int t = 0; t < TT; t++) {
    for (int i = tid; i < S_NUM * NUM_BSUB; i += 1024) Bf[i] = 0.0f;
    __syncthreads();
    const float* xrow = xw + (size_t)t * DD * BB;
    #pragma unroll 4
    for (int j = 0; j < E_NUM / 128; j++) {
      int e   = slot + j * 128;
      int idx = e * BB + cb;
      int   in_s  = ein[idx];
      int   out_s = eout[idx];
      int   pdf   = epdf[idx];
      float prob  = eprob[idx];
      float v = A[in_s * NUM_BSUB + b8] * prob * xrow[pdf * BB + cb];
      __hip_atomic_fetch_add(&Bf[out_s * NUM_BSUB + b8], v,
                             __ATOMIC_RELAXED, __HIP_MEMORY_SCOPE_WORKGROUP);
    }
    wait_ds0();
    __syncthreads();
    float p = 0.0f;
    #pragma unroll
    for (int k = 0; k < 4; k++) { int s = slot + k * 128; p += Bf[s * NUM_BSUB + b8]; }
    p += __shfl_xor(p, 8, 32);
    p += __shfl_xor(p, 16, 32);
    if (lane < 8) red[wv * 8 + lane] = p;
    __syncthreads();
    if (tid < 8) {
      float s = 0.0f;
      for (int w = 0; w < 32; w++) s += red[w * 8 + tid];
      asumv[tid] = s;
      if (t < len[tid]) logz[tid] += logf(s);
    }
    __syncthreads();
    {
      float as  = asumv[b8];
      float inv = 1.0f / as;
      bool  m   = (t < len[b8]);
      #pragma unroll
      for (int k = 0; k < 4; k++) {
        int s = slot + k * 128;
        int i = s * NUM_BSUB + b8;
        float ad = (Bf[i] + coeff * lk[i] * as) * inv;
        Bf[i] = m ? ad : A[i];
      }
    }
    __syncthreads();
    float* tmp = A; A = Bf; Bf = tmp;
  }
  float p = 0.0f;
  #pragma unroll
  for (int k = 0; k < 4; k++) {
    int s = slot + k * 128;
    p += A[s * NUM_BSUB + b8] * finalv[(size_t)cb * S_NUM + s];
  }
  p += __shfl_xor(p, 8, 32);
  p += __shfl_xor(p, 16, 32);
  if (lane < 8) red[wv * 8 + lane] = p;
  __syncthreads();
  if (tid < 8) {
    float s = 0.0f;
    for (int w = 0; w < 32; w++) s += red[w * 8 + tid];
    objf[g * NUM_BSUB + tid] = logz[tid] + logf(s);
  }
}

// ---------------- kernel 5: combine ----------------
__global__ void k_combine(const float* __restrict__ denp, const float* __restrict__ nump,
                          const int* __restrict__ lens, float* __restrict__ out) {
  int i = threadIdx.x;  // 32 threads = 1 wave
  float d = denp[i], n = nump[i];
  float L = (float)lens[i];
  for (int m = 16; m > 0; m >>= 1) {
    d += __shfl_xor(d, m, 32);
    n += __shfl_xor(n, m, 32);
    L += __shfl_xor(L, m, 32);
  }
  if (i == 0) out[0] = -(n - d) / L;
}

// ---------------- host launcher ----------------
extern "C" void kernel_launch(void* const* d_in, const int* in_sizes, int n_in,
                              void* d_out, int out_size, void* d_ws, size_t ws_size,
                              hipStream_t stream) {
  (void)in_sizes; (void)n_in; (void)out_size; (void)ws_size;
  const float* x         = (const float*)d_in[0];
  const int*   xlen      = (const int*)d_in[1];
  const int*   den_in    = (const int*)d_in[2];
  const int*   den_out   = (const int*)d_in[3];
  const int*   den_pdf   = (const int*)d_in[4];
  const float* den_prob  = (const float*)d_in[5];
  const float* den_leaky = (const float*)d_in[6];
  const float* den_final = (const float*)d_in[7];
  const int*   num_in    = (const int*)d_in[8];
  const int*   num_out   = (const int*)d_in[9];
  const int*   num_pdf   = (const int*)d_in[10];
  const float* num_prob  = (const float*)d_in[11];
  const float* num_leaky = (const float*)d_in[12];
  const float* num_final = (const float*)d_in[13];

  char* ws = (char*)d_ws;
  size_t off = 0;
  float* xw = (float*)(ws + off);  off += (size_t)TT * DD * BB * 4;      // 104.9 MB
  int*   tin  = (int*)(ws + off);  off += (size_t)E_NUM * BB * 4;
  int*   tout = (int*)(ws + off);  off += (size_t)E_NUM * BB * 4;
  int*   tpdf = (int*)(ws + off);  off += (size_t)E_NUM * BB * 4;
  float* tprob= (float*)(ws + off);off += (size_t)E_NUM * BB * 4;
  float* denp = (float*)(ws + off);off += 32 * 4;
  float* nump = (float*)(ws + off);off += 32 * 4;

  const size_t denLds = (size_t)DEN_LDS_FLOATS * 4;   // 88,960 B
  const size_t numLds = (size_t)NUM_LDS_FLOATS * 4;   // 50,272 B
  (void)hipFuncSetAttribute(reinterpret_cast<const void*>(&k_den),
                            hipFuncAttributeMaxDynamicSharedMemorySize, (int)denLds);
  (void)hipFuncSetAttribute(reinterpret_cast<const void*>(&k_num),
                            hipFuncAttributeMaxDynamicSharedMemorySize, (int)numLds);

  k_exp_tr<<<TT * (DD / 32), dim3(32, 32), 0, stream>>>(x, xw);
  k_tr_edges<<<(E_NUM * BB + 255) / 256, 256, 0, stream>>>(
      num_in, num_out, num_pdf, num_prob, tin, tout, tpdf, tprob);
  k_den<<<DEN_BLOCKS, 1024, denLds, stream>>>(
      xw, den_in, den_out, den_pdf, den_prob, den_leaky, den_final, xlen, denp);
  k_num<<<NUM_BLOCKS, 1024, numLds, stream>>>(
      xw, tin, tout, tpdf, tprob, num_leaky, num_final, xlen, nump);
  k_combine<<<1, 32, 0, stream>>>(denp, nump, xlen, (float*)d_out);
}